// RadialProbability_76819785056933
// MI455X (gfx1250) — compile-verified
//
#include <hip/hip_runtime.h>

typedef __attribute__((ext_vector_type(2))) float v2f;
typedef __attribute__((ext_vector_type(8))) float v8f;

#define N_PTS 12288
#define TILE 16
#define NTILES (N_PTS / TILE)      // 768 column tiles
#define WAVES_PER_BLOCK 8

// Compute one 16x16 tile of p = exp(-dist) for rows [rowBase,rowBase+16),
// cols [colBase,colBase+16).  `a` holds the 16x4 A tile (pre-scaled by 2),
// sqi[r] = ||x_row||^2 for this lane's 8 rows (C/D layout, ISA 7.12.2).
__device__ __forceinline__ v8f tile_prob(const float* __restrict__ x,
                                         int rowBase, int colBase,
                                         v2f a, const float* sqi, int lane) {
  const int cn  = lane & 15;
  const int col = colBase + cn;
  // Full float3 of this lane's column: feeds both the B tile and ||x_j||^2.
  const float x0 = x[col * 3 + 0];
  const float x1 = x[col * 3 + 1];
  const float x2 = x[col * 3 + 2];
  const float sj = x0 * x0 + x1 * x1 + x2 * x2;

  // B is 4x16 f32: lanes 0-15 carry K=0,1; lanes 16-31 carry K=2,3 (K=3 pad=0).
  v2f b;
  if (lane < 16) { b.x = x0; b.y = x1; }
  else           { b.x = x2; b.y = 0.0f; }

  // Fold the bias into C: D = (2A)xB + C = 2 x_i.x_j - ||x_i||^2 - ||x_j||^2 = -dist
  v8f c;
#pragma unroll
  for (int r = 0; r < 8; ++r) c[r] = -(sqi[r] + sj);

  v8f d = __builtin_amdgcn_wmma_f32_16x16x4_f32(
      /*neg_a=*/false, a, /*neg_b=*/false, b,
      /*c_mod=*/(short)0, c, /*reuse_a=*/false, /*reuse_b=*/false);

  const int mhalf = (lane >> 4) * 8;
  v8f p;
#pragma unroll
  for (int r = 0; r < 8; ++r) {
    float e = __expf(d[r]);                       // v_exp_f32 (TRANS)
    int row = rowBase + mhalf + r;
    p[r] = (row == col) ? 0.0f : e;               // diagonal: exp(-INF) -> 0
  }
  return p;
}

__global__ void __launch_bounds__(256)
radial_prob_kernel(const float* __restrict__ x, float* __restrict__ out) {
  __shared__ float rowsum[TILE];

  const int tid     = threadIdx.x;
  const int lane    = tid & 31;
  const int wave    = tid >> 5;
  const int rowBase = blockIdx.x * TILE;
  const int mhalf   = (lane >> 4) * 8;

  if (tid < TILE) rowsum[tid] = 0.0f;

  // A tile (16x4 f32, pre-scaled by 2): lane holds row rowBase+(lane&15),
  // lanes 0-15 -> K=0,1 ; lanes 16-31 -> K=2,3 (K=3 pad = 0).
  const int ar = rowBase + (lane & 15);
  const float a0 = x[ar * 3 + 0];
  const float a1 = x[ar * 3 + 1];
  const float a2 = x[ar * 3 + 2];
  v2f a;
  if (lane < 16) { a.x = 2.0f * a0; a.y = 2.0f * a1; }
  else           { a.x = 2.0f * a2; a.y = 0.0f; }

  // ||x_row||^2 for this lane's 8 rows of the C/D layout.
  float sqi[8];
#pragma unroll
  for (int r = 0; r < 8; ++r) {
    int row = rowBase + mhalf + r;
    float y0 = x[row * 3 + 0], y1 = x[row * 3 + 1], y2 = x[row * 3 + 2];
    sqi[r] = y0 * y0 + y1 * y1 + y2 * y2;
  }

  __syncthreads();

  // ---------- Phase 1: row sums of exp(-dist), no global writes ----------
  float acc[8];
#pragma unroll
  for (int r = 0; r < 8; ++r) acc[r] = 0.0f;

  for (int t = wave; t < NTILES; t += WAVES_PER_BLOCK) {
    v8f p = tile_prob(x, rowBase, t * TILE, a, sqi, lane);
#pragma unroll
    for (int r = 0; r < 8; ++r) acc[r] += p[r];
  }

  // Reduce across the 16 column-lanes of each half-wave (masks < 16 keep halves apart).
#pragma unroll
  for (int off = 1; off < 16; off <<= 1) {
#pragma unroll
    for (int r = 0; r < 8; ++r) acc[r] += __shfl_xor(acc[r], off, 32);
  }
  if ((lane & 15) == 0) {
#pragma unroll
    for (int r = 0; r < 8; ++r) atomicAdd(&rowsum[mhalf + r], acc[r]);  // ds_add_f32
  }
  __syncthreads();

  float inv[8];
#pragma unroll
  for (int r = 0; r < 8; ++r) inv[r] = 1.0f / rowsum[mhalf + r];

  // ---------- Phase 2: recompute tiles, single normalized write ----------
  for (int t = wave; t < NTILES; t += WAVES_PER_BLOCK) {
    const int colBase = t * TILE;
    v8f p = tile_prob(x, rowBase, colBase, a, sqi, lane);
    const int col = colBase + (lane & 15);
#pragma unroll
    for (int r = 0; r < 8; ++r) {
      size_t idx = (size_t)(rowBase + mhalf + r) * (size_t)N_PTS + (size_t)col;
      out[idx] = p[r] * inv[r];
    }
  }
}

extern "C" void kernel_launch(void* const* d_in, const int* in_sizes, int n_in,
                              void* d_out, int out_size, void* d_ws, size_t ws_size,
                              hipStream_t stream) {
  (void)in_sizes; (void)n_in; (void)d_ws; (void)ws_size; (void)out_size;
  const float* x = (const float*)d_in[0];
  float* out = (float*)d_out;
  radial_prob_kernel<<<NTILES, 256, 0, stream>>>(x, out);
}